// SelectiveSSM_15247133901132
// MI455X (gfx1250) — compile-verified
//
#include <hip/hip_runtime.h>
#include <hip/hip_bf16.h>
#include <math.h>

// Problem dims
#define BATCH   2
#define LSEQ    2048
#define DMODEL  768
#define DSTATE  16
#define DINNER  1536
#define DTRANK  48
#define KCONV   4
#define NDBLC   (DTRANK + 2 * DSTATE)   // 80

// Chunked scan config
#define CH 32          // number of chunks
#define CL 64          // chunk length (CH*CL == LSEQ)

typedef float v2f __attribute__((ext_vector_type(2)));
typedef float v8f __attribute__((ext_vector_type(8)));
typedef unsigned int uint32x4 __attribute__((ext_vector_type(4)));
typedef int int32x8 __attribute__((ext_vector_type(8)));
typedef int int32x4 __attribute__((ext_vector_type(4)));

// ---------------- fp32 WMMA GEMM: C[M,N] = act(A[M,K] * W[N,K]^T + bias) -------------
// Register-blocked: each wave owns a 32x32 output (2x2 WMMA accumulators), block = 64x128.
// flags bit0: softplus epilogue; bit1: TDM double-buffered staging (requires K%KT==0, N%BN==0, M%BM==0)
#define BM 64
#define BN 128
#define KT 32
#define STR 34   // padded LDS row stride (dwords) == TDM pad_interval 32dw + pad_amount 2dw

// Issue a 2-D TDM tile load: tile_d0 x tile_d1 elements (f32) from row-major tensor
// (row stride = stride0 elements) into LDS at byte offset lds_addr, with +2dword row padding.
// D# layout per CDNA5 ISA 8.3-8.5.
__device__ __forceinline__ void tdm_load_2d(unsigned lds_addr, const float* gaddr,
                                            unsigned tensor_d0, unsigned tensor_d1,
                                            unsigned long long stride0,
                                            unsigned tile_d0, unsigned tile_d1)
{
    unsigned long long ga = (unsigned long long)(uintptr_t)gaddr;
    uint32x4 g0 = {
        1u,                                                   // count=1 (valid user D#)
        lds_addr,                                             // LDS byte address
        (unsigned)ga,                                         // global_addr[31:0]
        (unsigned)((ga >> 32) & 0x01FFFFFFu) | (2u << 30)     // global_addr[56:32] | type=2
    };
    // word0: data_size=4B (2<<16) | pad_enable (1<<20) | pad_interval=32dw (4<<22) | pad_amount=2dw (1<<25)
    const unsigned w0 = (2u << 16) | (1u << 20) | (4u << 22) | (1u << 25);
    int32x8 g1 = {
        (int)w0,
        (int)((tensor_d0 & 0xFFFFu) << 16),                           // tensor_dim0[15:0] << 16
        (int)((tensor_d0 >> 16) | ((tensor_d1 & 0xFFFFu) << 16)),     // td0[31:16] | td1[15:0]<<16
        (int)((tensor_d1 >> 16) | (tile_d0 << 16)),                   // td1[31:16] | tile_dim0<<16
        (int)(tile_d1 & 0xFFFFu),                                     // tile_dim1 | tile_dim2(=0)<<16
        (int)(unsigned)stride0,                                       // tensor_dim0_stride[31:0]
        (int)(unsigned)(stride0 >> 32),                               // stride0[47:32] | stride1[15:0]<<16
        0                                                             // stride1[47:16]
    };
    int32x4 z4 = {0, 0, 0, 0};
    int32x8 z8 = {0, 0, 0, 0, 0, 0, 0, 0};
    __builtin_amdgcn_tensor_load_to_lds(g0, g1, z4, z4, z8, 0);
}

__global__ __launch_bounds__(256)
void wmma_gemm_f32(const float* __restrict__ A, const float* __restrict__ W,
                   const float* __restrict__ bias, float* __restrict__ C,
                   int M, int N, int K, int lda, int ldc, int flags)
{
    // double-buffered combined LDS: [buf][ A: BM*STR | W: BN*STR ]  -> 52.2 KB
    __shared__ float smem[2 * (BM + BN) * STR];

    const int tid  = threadIdx.x;
    const int lane = tid & 31;
    const int wave = tid >> 5;
    const int wr   = wave >> 2;   // 0..1 : 32-row slab
    const int wc   = wave & 3;    // 0..3 : 32-col slab
    const int half = lane >> 4;
    const int lr   = lane & 15;

    const int m0 = blockIdx.y * BM;
    const int n0 = blockIdx.x * BN;
    const bool use_tdm = (flags & 2) != 0;

    v8f acc00 = {0,0,0,0,0,0,0,0}, acc01 = {0,0,0,0,0,0,0,0};
    v8f acc10 = {0,0,0,0,0,0,0,0}, acc11 = {0,0,0,0,0,0,0,0};

    const int nst = (K + KT - 1) / KT;

    for (int st = 0; st < nst; ++st) {
        const int k0 = st * KT;
        const float* Ab;
        const float* Wb;

        if (use_tdm) {
            // ---- TDM double-buffered staging: DMA overlaps WMMA of previous stage ----
            const unsigned bufA = (unsigned)((st & 1) * (BM + BN) * STR) * 4u;
            if (wave == 0) {
                if (st == 0) {   // prologue: issue stage 0
                    tdm_load_2d(bufA,                 A + (size_t)m0 * lda + k0, K, M, (unsigned long long)lda, KT, BM);
                    tdm_load_2d(bufA + BM * STR * 4u, W + (size_t)n0 * K   + k0, K, N, (unsigned long long)K,   KT, BN);
                }
                __builtin_amdgcn_s_wait_tensorcnt(0);   // stage st resident
            }
            __syncthreads();                             // publish buf, and all waves done with buf^1
            if (wave == 0 && st + 1 < nst) {             // issue stage st+1 into other buffer
                const unsigned nbufA = (unsigned)(((st + 1) & 1) * (BM + BN) * STR) * 4u;
                const int k1 = k0 + KT;
                tdm_load_2d(nbufA,                 A + (size_t)m0 * lda + k1, K, M, (unsigned long long)lda, KT, BM);
                tdm_load_2d(nbufA + BM * STR * 4u, W + (size_t)n0 * K   + k1, K, N, (unsigned long long)K,   KT, BN);
            }
            Ab = smem + (st & 1) * (BM + BN) * STR;
            Wb = Ab + BM * STR;
        } else {
            // ---- manual guarded staging (edge GEMMs: N=80 or K=48) into buffer 0 ----
            #pragma unroll
            for (int q = 0; q < 2; ++q) {
                int f   = tid + (q << 8);
                int row = f >> 3;
                int c4  = (f & 7) << 2;
                int gm  = m0 + row;
                int gk  = k0 + c4;
                float4 v = make_float4(0.f, 0.f, 0.f, 0.f);
                if (gm < M) {
                    if (gk + 3 < K) {
                        v = *(const float4*)(A + (size_t)gm * lda + gk);
                    } else {
                        float* vp = (float*)&v;
                        #pragma unroll
                        for (int j = 0; j < 4; ++j)
                            if (gk + j < K) vp[j] = A[(size_t)gm * lda + gk + j];
                    }
                }
                const float* vp = (const float*)&v;
                smem[row * STR + c4 + 0] = vp[0];
                smem[row * STR + c4 + 1] = vp[1];
                smem[row * STR + c4 + 2] = vp[2];
                smem[row * STR + c4 + 3] = vp[3];
            }
            #pragma unroll
            for (int q = 0; q < 4; ++q) {
                int f   = tid + (q << 8);
                int row = f >> 3;
                int c4  = (f & 7) << 2;
                int gn  = n0 + row;
                int gk  = k0 + c4;
                float4 v = make_float4(0.f, 0.f, 0.f, 0.f);
                if (gn < N) {
                    if (gk + 3 < K) {
                        v = *(const float4*)(W + (size_t)gn * K + gk);
                    } else {
                        float* vp = (float*)&v;
                        #pragma unroll
                        for (int j = 0; j < 4; ++j)
                            if (gk + j < K) vp[j] = W[(size_t)gn * K + gk + j];
                    }
                }
                const float* vp = (const float*)&v;
                smem[BM * STR + row * STR + c4 + 0] = vp[0];
                smem[BM * STR + row * STR + c4 + 1] = vp[1];
                smem[BM * STR + row * STR + c4 + 2] = vp[2];
                smem[BM * STR + row * STR + c4 + 3] = vp[3];
            }
            __syncthreads();
            Ab = smem;
            Wb = smem + BM * STR;
        }

        // ---- 8 k-steps x 4 WMMA (2x2 register tile) over the KT slab ----
        const float* ap0 = &Ab[(wr * 32 +      lr) * STR + 2 * half];
        const float* ap1 = &Ab[(wr * 32 + 16 + lr) * STR + 2 * half];
        const float* bp0 = &Wb[(wc * 32 +      lr) * STR + 2 * half];
        const float* bp1 = &Wb[(wc * 32 + 16 + lr) * STR + 2 * half];
        #pragma unroll
        for (int kk = 0; kk < KT; kk += 4) {
            v2f a0; a0.x = ap0[kk]; a0.y = ap0[kk + 1];
            v2f a1; a1.x = ap1[kk]; a1.y = ap1[kk + 1];
            v2f b0; b0.x = bp0[kk]; b0.y = bp0[kk + 1];
            v2f b1; b1.x = bp1[kk]; b1.y = bp1[kk + 1];
            acc00 = __builtin_amdgcn_wmma_f32_16x16x4_f32(false, a0, false, b0, (short)0, acc00, false, false);
            acc01 = __builtin_amdgcn_wmma_f32_16x16x4_f32(false, a0, false, b1, (short)0, acc01, false, false);
            acc10 = __builtin_amdgcn_wmma_f32_16x16x4_f32(false, a1, false, b0, (short)0, acc10, false, false);
            acc11 = __builtin_amdgcn_wmma_f32_16x16x4_f32(false, a1, false, b1, (short)0, acc11, false, false);
        }
        if (!use_tdm) __syncthreads();
    }

    // ---- epilogue ----
    #pragma unroll
    for (int mi = 0; mi < 2; ++mi) {
        const int mb = m0 + wr * 32 + mi * 16 + 8 * half;
        #pragma unroll
        for (int ni = 0; ni < 2; ++ni) {
            const int n = n0 + wc * 32 + ni * 16 + lr;
            if (n >= N) continue;
            const v8f* accp = (mi == 0) ? (ni == 0 ? &acc00 : &acc01)
                                        : (ni == 0 ? &acc10 : &acc11);
            const float bv = bias ? bias[n] : 0.f;
            #pragma unroll
            for (int r = 0; r < 8; ++r) {
                const int m = mb + r;
                if (m < M) {
                    float v = (*accp)[r] + bv;
                    if (flags & 1) v = (v > 20.f) ? v : log1pf(__expf(v));  // softplus
                    C[(size_t)m * ldc + n] = v;
                }
            }
        }
    }
}

// ---------------- depthwise causal conv1d (K=4) + bias + SiLU ----------------
__global__ __launch_bounds__(256)
void conv_silu_kernel(const float* __restrict__ xz, const float* __restrict__ cw,
                      const float* __restrict__ cb, float* __restrict__ xc)
{
    const int i = blockIdx.x * 256 + threadIdx.x;     // over BATCH*LSEQ*DINNER
    const int d  = i % DINNER;
    const int bl = i / DINNER;
    const int l  = bl % LSEQ;
    const int b  = bl / LSEQ;

    float acc = cb[d];
    #pragma unroll
    for (int k = 0; k < KCONV; ++k) {
        const int ls = l - (KCONV - 1) + k;
        if (ls >= 0)
            acc += xz[((size_t)b * LSEQ + ls) * (2 * DINNER) + d] * cw[d * KCONV + k];
    }
    const float sig = 1.f / (1.f + __expf(-acc));
    xc[(size_t)bl * DINNER + d] = acc * sig;
}

// ---------------- chunked selective scan ----------------
// h[l] = exp(dt[l]*A[s]) * h[l-1] + dt[l]*xc[l]*B[l][s];  chunk decay = exp(A[s]*sum(dt)).

// Pass 1: per-chunk local scan with h_in = 0 -> h_end0[b,c,d,s], sumdt[b,c,d]
__global__ __launch_bounds__(256)
void scan_chunk_pass1(const float* __restrict__ xc, const float* __restrict__ xdbl,
                      const float* __restrict__ dt, const float* __restrict__ A_log,
                      float* __restrict__ hend0, float* __restrict__ sumdt)
{
    const int b = blockIdx.x;
    const int c = blockIdx.y;
    const int d = blockIdx.z * 256 + threadIdx.x;

    __shared__ float Bs[CL * DSTATE];   // 4 KB: B vectors for the whole chunk
    for (int i = threadIdx.x; i < CL * DSTATE; i += 256) {
        const int ll = i >> 4, s = i & 15;
        const size_t row = (size_t)b * LSEQ + c * CL + ll;
        Bs[i] = xdbl[row * NDBLC + DTRANK + s];
    }
    __syncthreads();

    float Arow[DSTATE], h[DSTATE];
    #pragma unroll
    for (int s = 0; s < DSTATE; ++s) {
        Arow[s] = -__expf(A_log[(size_t)d * DSTATE + s]);
        h[s] = 0.f;
    }
    float sdt = 0.f;

    for (int ll = 0; ll < CL; ++ll) {
        const size_t row = (size_t)b * LSEQ + c * CL + ll;
        const size_t idx = row * DINNER + d;
        const float dtv = dt[idx];
        const float xcv = xc[idx];
        const float dtx = dtv * xcv;
        sdt += dtv;
        const float* Bl = &Bs[ll * DSTATE];
        #pragma unroll
        for (int s = 0; s < DSTATE; ++s)
            h[s] = __expf(dtv * Arow[s]) * h[s] + dtx * Bl[s];
    }

    const size_t base = (size_t)(b * CH + c) * DINNER + d;
    #pragma unroll
    for (int s = 0; s < DSTATE; ++s) hend0[base * DSTATE + s] = h[s];
    sumdt[base] = sdt;
}

// Pass 2: sequential combine over chunks per (b,d,s) -> h_in[b,c,d,s]
__global__ __launch_bounds__(256)
void scan_combine(const float* __restrict__ hend0, const float* __restrict__ sumdt,
                  const float* __restrict__ A_log, float* __restrict__ hin)
{
    const int i = blockIdx.x * 256 + threadIdx.x;     // over BATCH*DINNER*DSTATE
    const int s  = i & 15;
    const int dd = (i >> 4) % DINNER;
    const int b  = i / (DINNER * DSTATE);

    const float Ads = -__expf(A_log[(size_t)dd * DSTATE + s]);
    float h = 0.f;
    for (int c = 0; c < CH; ++c) {
        const size_t base = (size_t)(b * CH + c) * DINNER + dd;
        hin[base * DSTATE + s] = h;
        h = __expf(Ads * sumdt[base]) * h + hend0[base * DSTATE + s];
    }
}

// Pass 3: re-run each chunk from correct h_in; fuse y = <h,C> + xc*D, y *= silu(z). In place over dt.
__global__ __launch_bounds__(256)
void scan_chunk_pass2(const float* __restrict__ xz, const float* __restrict__ xc,
                      const float* __restrict__ xdbl, const float* __restrict__ A_log,
                      const float* __restrict__ D_param, const float* __restrict__ hin,
                      float* dt_y)
{
    const int b = blockIdx.x;
    const int c = blockIdx.y;
    const int d = blockIdx.z * 256 + threadIdx.x;

    __shared__ float BCs[CL * 2 * DSTATE];  // 8 KB: B and C vectors for the chunk
    for (int i = threadIdx.x; i < CL * 2 * DSTATE; i += 256) {
        const int ll = i >> 5, col = i & 31;
        const size_t row = (size_t)b * LSEQ + c * CL + ll;
        BCs[i] = xdbl[row * NDBLC + DTRANK + col];
    }
    __syncthreads();

    float Arow[DSTATE], h[DSTATE];
    const size_t base = (size_t)(b * CH + c) * DINNER + d;
    #pragma unroll
    for (int s = 0; s < DSTATE; ++s) {
        Arow[s] = -__expf(A_log[(size_t)d * DSTATE + s]);
        h[s] = hin[base * DSTATE + s];
    }
    const float Dv = D_param[d];

    for (int ll = 0; ll < CL; ++ll) {
        const size_t row = (size_t)b * LSEQ + c * CL + ll;
        const size_t idx = row * DINNER + d;
        const float dtv = dt_y[idx];
        const float xcv = xc[idx];
        const float zv  = xz[row * (2 * DINNER) + DINNER + d];
        const float dtx = dtv * xcv;
        const float* Bl = &BCs[ll * 2 * DSTATE];
        const float* Cl = Bl + DSTATE;

        float y = 0.f;
        #pragma unroll
        for (int s = 0; s < DSTATE; ++s) {
            h[s] = __expf(dtv * Arow[s]) * h[s] + dtx * Bl[s];
            y += h[s] * Cl[s];
        }
        y += xcv * Dv;
        const float sig = 1.f / (1.f + __expf(-zv));
        dt_y[idx] = y * (zv * sig);
    }
}

// ---------------- launch ----------------
extern "C" void kernel_launch(void* const* d_in, const int* in_sizes, int n_in,
                              void* d_out, int out_size, void* d_ws, size_t ws_size,
                              hipStream_t stream)
{
    (void)in_sizes; (void)n_in; (void)out_size; (void)ws_size;

    const float* x          = (const float*)d_in[0];
    const float* in_proj_w  = (const float*)d_in[1];
    const float* conv_w     = (const float*)d_in[2];
    const float* conv_b     = (const float*)d_in[3];
    const float* A_log      = (const float*)d_in[4];
    const float* D_param    = (const float*)d_in[5];
    const float* x_proj_w   = (const float*)d_in[6];
    const float* dt_proj_w  = (const float*)d_in[7];
    const float* dt_proj_b  = (const float*)d_in[8];
    const float* out_proj_w = (const float*)d_in[9];

    const int M = BATCH * LSEQ;                 // 4096 rows

    float* ws    = (float*)d_ws;
    float* xz    = ws;                                       // M * 3072
    float* xc    = xz    + (size_t)M * 2 * DINNER;           // M * 1536
    float* xdbl  = xc    + (size_t)M * DINNER;               // M * 80
    float* dt_y  = xdbl  + (size_t)M * NDBLC;                // M * 1536 (dt, then y in place)
    float* hend0 = dt_y  + (size_t)M * DINNER;               // B*CH*DINNER*16
    float* sumdt = hend0 + (size_t)BATCH * CH * DINNER * DSTATE;  // B*CH*DINNER
    float* hin   = sumdt + (size_t)BATCH * CH * DINNER;      // B*CH*DINNER*16

    dim3 blk(256);

    // 1) in_proj (TDM path: K=768, N=3072 both tile-exact)
    wmma_gemm_f32<<<dim3((2 * DINNER) / BN, M / BM), blk, 0, stream>>>(
        x, in_proj_w, nullptr, xz, M, 2 * DINNER, DMODEL, DMODEL, 2 * DINNER, 2);

    // 2) depthwise causal conv + SiLU -> xc[M,1536]
    conv_silu_kernel<<<(M * DINNER) / 256, blk, 0, stream>>>(xz, conv_w, conv_b, xc);

    // 3) x_proj (manual path: N=80 edge)
    wmma_gemm_f32<<<dim3((NDBLC + BN - 1) / BN, M / BM), blk, 0, stream>>>(
        xc, x_proj_w, nullptr, xdbl, M, NDBLC, DINNER, DINNER, NDBLC, 0);

    // 4) dt = softplus(x_dbl[:, :48] @ dt_proj_w^T + b)  (manual path: K=48 edge)
    wmma_gemm_f32<<<dim3(DINNER / BN, M / BM), blk, 0, stream>>>(
        xdbl, dt_proj_w, dt_proj_b, dt_y, M, DINNER, DTRANK, NDBLC, DINNER, 1);

    // 5) chunked selective scan (3 passes; y overwrites dt buffer)
    scan_chunk_pass1<<<dim3(BATCH, CH, DINNER / 256), blk, 0, stream>>>(
        xc, xdbl, dt_y, A_log, hend0, sumdt);
    scan_combine<<<(BATCH * DINNER * DSTATE) / 256, blk, 0, stream>>>(
        hend0, sumdt, A_log, hin);
    scan_chunk_pass2<<<dim3(BATCH, CH, DINNER / 256), blk, 0, stream>>>(
        xz, xc, xdbl, A_log, D_param, hin, dt_y);

    // 6) out_proj (TDM path: K=1536, N=768 both tile-exact)
    wmma_gemm_f32<<<dim3(DMODEL / BN, M / BM), blk, 0, stream>>>(
        dt_y, out_proj_w, nullptr, (float*)d_out, M, DMODEL, DINNER, DINNER, DMODEL, 2);
}